// Network_21998822490747
// MI455X (gfx1250) — compile-verified
//
#include <hip/hip_runtime.h>
#include <hip/hip_bf16.h>

// Problem constants (match reference)
constexpr int Tseq = 4096;
constexpr int Edim = 256;
constexpr int Hdim = 10;
constexpr int Odim = 50257;
constexpr int Hpad = 12;                        // K padded to 3 WMMA k-chunks of 4
constexpr int OtilesPerWave = 8;                // 128 output cols per wave
constexpr int Ogroups = (Odim + 16 * OtilesPerWave - 1) / (16 * OtilesPerWave); // 393
constexpr int Opad = Ogroups * 16 * OtilesPerWave;                              // 50304

typedef float v2f __attribute__((ext_vector_type(2)));
typedef float v8f __attribute__((ext_vector_type(8)));

// Workspace layout (bytes)
constexpr size_t WS_XGT    = 0;                                         // 40*T f32
constexpr size_t WS_HSPAD  = WS_XGT   + (size_t)4 * Hdim * Tseq * 4;    // T*12 f32
constexpr size_t WS_WPAD   = WS_HSPAD + (size_t)Tseq * Hpad * 4;        // Opad*12 f32
constexpr size_t WS_BPAD   = WS_WPAD  + (size_t)Opad * Hpad * 4;        // Opad f32

// ---------------------------------------------------------------------------
// Kernel 0: zero-pad W_out -> W_pad[Opad][12], b_out -> b_pad[Opad].
// Makes every A/B load in the WMMA kernel unconditional (EXEC stays all-1s).
// ---------------------------------------------------------------------------
__global__ void k_pad_weights(const float* __restrict__ W_out,
                              const float* __restrict__ b_out,
                              float* __restrict__ W_pad,
                              float* __restrict__ b_pad) {
  int i = blockIdx.x * 256 + threadIdx.x;
  if (i < Opad) {
    bool v = (i < Odim);
    b_pad[i] = v ? b_out[i] : 0.0f;
#pragma unroll
    for (int k = 0; k < Hpad; ++k)
      W_pad[(size_t)i * Hpad + k] =
          (v && k < Hdim) ? W_out[(size_t)i * Hdim + k] : 0.0f;
  }
}

// ---------------------------------------------------------------------------
// Kernel 1: embedding gather + xg = emb @ w_ih^T + (b_ih + b_hh),
// written TRANSPOSED as xgT[40][T] so the sequential kernel can stream
// contiguous per-gate columns with vector loads. One block (64 thr) per t.
// ---------------------------------------------------------------------------
__global__ void k_embed_gates(const int* __restrict__ x,
                              const float* __restrict__ emb,
                              const float* __restrict__ w_ih,
                              const float* __restrict__ b_ih,
                              const float* __restrict__ b_hh,
                              float* __restrict__ xgT) {
  __shared__ __align__(16) float erow[Edim];
  const int t   = blockIdx.x;
  const int tid = threadIdx.x;                 // 64 threads
  const int row = x[t];
  ((float4*)erow)[tid] = ((const float4*)(emb + (size_t)row * Edim))[tid];
  __syncthreads();
  if (tid < 4 * Hdim) {                        // 40 gate dot-products
    float acc = b_ih[tid] + b_hh[tid];
    const float4* w  = (const float4*)(w_ih + (size_t)tid * Edim);
    const float4* e4 = (const float4*)erow;
#pragma unroll 8
    for (int i = 0; i < Edim / 4; ++i) {
      float4 a = e4[i], b = w[i];
      acc = fmaf(a.x, b.x, acc);
      acc = fmaf(a.y, b.y, acc);
      acc = fmaf(a.z, b.z, acc);
      acc = fmaf(a.w, b.w, acc);
    }
    xgT[(size_t)tid * Tseq + t] = acc;
  }
}

// ---------------------------------------------------------------------------
// Kernel 2: sequential LSTM scan. ONE wave32. Lane j owns gate columns
// {j, j+10, j+20, j+30} (mod-10 replicated for lanes 10..31). h broadcast
// through LDS each step (DS ops are in-order within a wave -> s_wait_dscnt).
// xg streams double-buffered in 8-step blocks to hide L2 latency.
// Emits hs padded to stride 12 (k=10,11 zero) for the WMMA projection.
// ---------------------------------------------------------------------------
__device__ __forceinline__ float sigm_(float v) {
  float e = __expf(-v);
  return __builtin_amdgcn_rcpf(1.0f + e);
}
__device__ __forceinline__ float tanh_(float v) {
  return fmaf(2.0f, sigm_(2.0f * v), -1.0f);   // saturates cleanly at +-1
}
__device__ __forceinline__ void load8(const float* __restrict__ p, float* dst) {
  float4 a = *(const float4*)(p);
  float4 b = *(const float4*)(p + 4);
  dst[0] = a.x; dst[1] = a.y; dst[2] = a.z; dst[3] = a.w;
  dst[4] = b.x; dst[5] = b.y; dst[6] = b.z; dst[7] = b.w;
}

__global__ void k_lstm_scan(const float* __restrict__ xgT,
                            const float* __restrict__ w_hh,
                            float* __restrict__ hs_pad) {
  __shared__ __align__(16) float lds_h[16];
  const int lane = threadIdx.x;          // 32 lanes, single wave
  const int jj   = lane % Hdim;

  float w0[Hdim], w1[Hdim], w2[Hdim], w3[Hdim];
#pragma unroll
  for (int k = 0; k < Hdim; ++k) {
    w0[k] = w_hh[(0 * Hdim + jj) * Hdim + k];
    w1[k] = w_hh[(1 * Hdim + jj) * Hdim + k];
    w2[k] = w_hh[(2 * Hdim + jj) * Hdim + k];
    w3[k] = w_hh[(3 * Hdim + jj) * Hdim + k];
  }

  const float* p0 = xgT + (size_t)(0 * Hdim + jj) * Tseq;
  const float* p1 = xgT + (size_t)(1 * Hdim + jj) * Tseq;
  const float* p2 = xgT + (size_t)(2 * Hdim + jj) * Tseq;
  const float* p3 = xgT + (size_t)(3 * Hdim + jj) * Tseq;

  float h[Hdim];
#pragma unroll
  for (int k = 0; k < Hdim; ++k) h[k] = 0.0f;
  float c = 0.0f;

  constexpr int BS = 8, NB = Tseq / BS;
  float cur0[BS], cur1[BS], cur2[BS], cur3[BS];
  float nxt0[BS], nxt1[BS], nxt2[BS], nxt3[BS];
  load8(p0, cur0); load8(p1, cur1); load8(p2, cur2); load8(p3, cur3);

  for (int b = 0; b < NB; ++b) {
    const int t0 = b * BS;
    if (b + 1 < NB) {                    // prefetch next block (hide latency)
      load8(p0 + t0 + BS, nxt0); load8(p1 + t0 + BS, nxt1);
      load8(p2 + t0 + BS, nxt2); load8(p3 + t0 + BS, nxt3);
    }
#pragma unroll
    for (int u = 0; u < BS; ++u) {
      float gi = cur0[u], gf = cur1[u], gg = cur2[u], go = cur3[u];
#pragma unroll
      for (int k = 0; k < Hdim; ++k) {
        gi = fmaf(w0[k], h[k], gi);
        gf = fmaf(w1[k], h[k], gf);
        gg = fmaf(w2[k], h[k], gg);
        go = fmaf(w3[k], h[k], go);
      }
      float iv = sigm_(gi), fv = sigm_(gf), gv = tanh_(gg), ov = sigm_(go);
      c = fmaf(fv, c, iv * gv);
      float hn = ov * tanh_(c);
      if (lane < Hpad) {
        float v = (lane < Hdim) ? hn : 0.0f;
        lds_h[lane] = v;
        hs_pad[(size_t)(t0 + u) * Hpad + lane] = v;  // zero-padded hs row
      }
      asm volatile("s_wait_dscnt 0" ::: "memory");   // DS in-order in a wave
#pragma unroll
      for (int k = 0; k < Hdim; ++k) h[k] = lds_h[k];
    }
#pragma unroll
    for (int k = 0; k < BS; ++k) {
      cur0[k] = nxt0[k]; cur1[k] = nxt1[k];
      cur2[k] = nxt2[k]; cur3[k] = nxt3[k];
    }
  }
}

// ---------------------------------------------------------------------------
// Kernel 3: logits = hs @ W_out^T + b_out via V_WMMA_F32_16X16X4_F32.
// One wave per 16(M) x 128(N) tile: A loaded once (8x reuse), 24 WMMAs,
// all loads unconditional (padded operands) so EXEC stays all-1s.
// Store path: uniform branch selects a branch-free NT-store stream for the
// 392/393 fully-in-range block columns (512B contiguous per half-wave row).
// ISA layouts (f32 16x16x4): lane L -> A[m=L&15][k=2*(L>>4)+v],
//                            B[k=2*(L>>4)+v][n=L&15], D[r]: m=r+8*(L>>4), n=L&15.
// ---------------------------------------------------------------------------
__global__ void k_logits_wmma(const float* __restrict__ hs_pad,
                              const float* __restrict__ W_pad,
                              const float* __restrict__ b_pad,
                              float* __restrict__ out) {
  const int lane  = threadIdx.x;       // 32
  const int half  = lane >> 4;
  const int l16   = lane & 15;
  const int tbase = blockIdx.y * 16;
  const int og0   = blockIdx.x * (16 * OtilesPerWave);

  // A tile (shared by all 8 sub-tiles): k-chunks {0,4,8} + 2*half
  const float* Arow = hs_pad + (size_t)(tbase + l16) * Hpad + 2 * half;
  v2f a0 = *(const v2f*)(Arow + 0);
  v2f a1 = *(const v2f*)(Arow + 4);
  v2f a2 = *(const v2f*)(Arow + 8);

  v8f acc[OtilesPerWave];
#pragma unroll
  for (int s = 0; s < OtilesPerWave; ++s) {
    const int o = og0 + s * 16 + l16;
    const float bo = b_pad[o];
    acc[s] = {bo, bo, bo, bo, bo, bo, bo, bo};
    const float* Bp = W_pad + (size_t)o * Hpad + 2 * half;
    v2f b0 = *(const v2f*)(Bp + 0);
    v2f b1 = *(const v2f*)(Bp + 4);
    v2f b2 = *(const v2f*)(Bp + 8);
    acc[s] = __builtin_amdgcn_wmma_f32_16x16x4_f32(false, a0, false, b0,
                                                   (short)0, acc[s], false, false);
    acc[s] = __builtin_amdgcn_wmma_f32_16x16x4_f32(false, a1, false, b1,
                                                   (short)0, acc[s], false, false);
    acc[s] = __builtin_amdgcn_wmma_f32_16x16x4_f32(false, a2, false, b2,
                                                   (short)0, acc[s], false, false);
  }

  // Uniform per-block predicate: scalar branch, keeps the hot store path
  // completely free of per-lane exec masking.
  const bool fullTile = (og0 + 16 * OtilesPerWave) <= Odim;
  if (fullTile) {
#pragma unroll
    for (int r = 0; r < 8; ++r) {
      const int m = tbase + r + 8 * half;
      float* rowp = out + (size_t)m * Odim + og0 + l16;
#pragma unroll
      for (int s = 0; s < OtilesPerWave; ++s)
        __builtin_nontemporal_store(acc[s][r], rowp + s * 16);
    }
  } else {
#pragma unroll
    for (int r = 0; r < 8; ++r) {
      const int m = tbase + r + 8 * half;
      float* rowp = out + (size_t)m * Odim;
#pragma unroll
      for (int s = 0; s < OtilesPerWave; ++s) {
        const int o = og0 + s * 16 + l16;
        if (o < Odim)
          __builtin_nontemporal_store(acc[s][r], rowp + o);
      }
    }
  }
}

// ---------------------------------------------------------------------------
extern "C" void kernel_launch(void* const* d_in, const int* in_sizes, int n_in,
                              void* d_out, int out_size, void* d_ws, size_t ws_size,
                              hipStream_t stream) {
  const int*   x     = (const int*)d_in[0];
  const float* emb   = (const float*)d_in[1];
  const float* w_ih  = (const float*)d_in[2];
  const float* w_hh  = (const float*)d_in[3];
  const float* b_ih  = (const float*)d_in[4];
  const float* b_hh  = (const float*)d_in[5];
  const float* W_out = (const float*)d_in[6];
  const float* b_out = (const float*)d_in[7];
  float* out = (float*)d_out;

  float* xgT    = (float*)((char*)d_ws + WS_XGT);
  float* hs_pad = (float*)((char*)d_ws + WS_HSPAD);
  float* W_pad  = (float*)((char*)d_ws + WS_WPAD);
  float* b_pad  = (float*)((char*)d_ws + WS_BPAD);

  k_pad_weights<<<(Opad + 255) / 256, 256, 0, stream>>>(W_out, b_out, W_pad, b_pad);
  k_embed_gates<<<Tseq, 64, 0, stream>>>(x, emb, w_ih, b_ih, b_hh, xgT);
  k_lstm_scan<<<1, 32, 0, stream>>>(xgT, w_hh, hs_pad);
  dim3 g3(Ogroups, Tseq / 16);
  k_logits_wmma<<<g3, 32, 0, stream>>>(hs_pad, W_pad, b_pad, out);
}